// ResamplingModel_24945170055314
// MI455X (gfx1250) — compile-verified
//
#include <hip/hip_runtime.h>
#include <math.h>

#define BB    512
#define NN    16384
#define TPB   256
#define ALPHA_F 0.5f
#define CU    (0.5f / 16384.0f)      // (1-alpha)/N
#define PADN  (NN + NN / 64)         // 16640 floats, bank-conflict-free padding
#define HALF_COUNT 4194304u          // (B*N)/2 for threefry counter split

typedef float v2f __attribute__((ext_vector_type(2)));
typedef float v8f __attribute__((ext_vector_type(8)));

__device__ __forceinline__ unsigned rotl32(unsigned x, unsigned r) {
    return (x << r) | (x >> (32u - r));
}

// Bit-exact JAX threefry2x32 for key = jax.random.key(1) -> (k0,k1) = (0,1)
__device__ __forceinline__ unsigned threefry2x32_k01(unsigned c0, unsigned c1, bool first) {
    const unsigned K0 = 0u, K1 = 1u;
    const unsigned K2 = 0x1BD11BDAu ^ K0 ^ K1;
    const unsigned ks[3] = {K0, K1, K2};
    const unsigned RA[4] = {13u, 15u, 26u, 6u};
    const unsigned RB[4] = {17u, 29u, 16u, 24u};
    unsigned x0 = c0 + K0, x1 = c1 + K1;
#pragma unroll
    for (int g = 0; g < 5; ++g) {
#pragma unroll
        for (int i = 0; i < 4; ++i) {
            unsigned r = (g & 1) ? RB[i] : RA[i];
            x0 += x1;
            x1 = rotl32(x1, r);
            x1 ^= x0;
        }
        x0 += ks[(g + 1) % 3];
        x1 += ks[(g + 2) % 3] + (unsigned)(g + 1);
    }
    return first ? x0 : x1;
}

// ---------------- Kernel A: per-row streaming logsumexp -------------------
__global__ __launch_bounds__(TPB) void k_row_lse(const float* __restrict__ pw,
                                                 float* __restrict__ lse) {
    __shared__ float sm[8], ss[8];
    const int b = blockIdx.x, tid = threadIdx.x;
    const float4* row4 = (const float4*)(pw + (size_t)b * NN);
    float m = -INFINITY, s = 0.0f;
    for (int g = tid; g < NN / 4; g += TPB) {
        float4 x = row4[g];
        float xm = fmaxf(fmaxf(x.x, x.y), fmaxf(x.z, x.w));
        float nm = fmaxf(m, xm);
        s = s * __expf(m - nm) + __expf(x.x - nm) + __expf(x.y - nm) +
            __expf(x.z - nm) + __expf(x.w - nm);
        m = nm;
    }
    // wave32 reduction of (max, sumexp) pairs
#pragma unroll
    for (int d = 16; d > 0; d >>= 1) {
        float om = __shfl_xor(m, d, 32);
        float os = __shfl_xor(s, d, 32);
        float nm = fmaxf(m, om);
        s = s * __expf(m - nm) + os * __expf(om - nm);
        m = nm;
    }
    if ((tid & 31) == 0) { sm[tid >> 5] = m; ss[tid >> 5] = s; }
    __syncthreads();
    if (tid == 0) {
        float M = sm[0], S = ss[0];
#pragma unroll
        for (int i = 1; i < 8; ++i) {
            float nm = fmaxf(M, sm[i]);
            S = S * __expf(M - nm) + ss[i] * __expf(sm[i] - nm);
            M = nm;
        }
        lse[b] = M + __logf(S);
    }
}

// ------------- Kernel B: transform + WMMA scan + sample + gather ----------
__global__ __launch_bounds__(TPB) void k_resample(const float* __restrict__ states,
                                                  const float* __restrict__ pw,
                                                  const float* __restrict__ lseP,
                                                  float* __restrict__ out_states,
                                                  float* __restrict__ out_w) {
    extern __shared__ float smem[];
    float* cdf    = smem;              // PADN floats (padded: f = e + (e>>6))
    float* totals = smem + PADN;       // 256 per-thread partial sums
    float* totsc  = totals + TPB;      // 256 inclusive-scanned partials
    float* sT     = totsc + TPB;       // [T, logT]

    const int b = blockIdx.x, tid = threadIdx.x;
    const float lse = lseP[b];
    const float4* row4 = (const float4*)(pw + (size_t)b * NN);

    // Phase 0: coalesced 128-bit loads, transform to q-probabilities in LDS
#pragma unroll
    for (int i = 0; i < NN / 4 / TPB; ++i) {          // 16 iterations
        int g = i * TPB + tid;
        float4 w = row4[g];
        int f = 4 * g + (g >> 4);                     // pad = (4g)>>6 = g>>4
        cdf[f + 0] = ALPHA_F * __expf(w.x - lse) + CU;
        cdf[f + 1] = ALPHA_F * __expf(w.y - lse) + CU;
        cdf[f + 2] = ALPHA_F * __expf(w.z - lse) + CU;
        cdf[f + 3] = ALPHA_F * __expf(w.w - lse) + CU;
    }
    __syncthreads();

    // Phase 1: per-thread serial scan of 64 contiguous elems (conflict-free)
    {
        float run = 0.0f;
        int base = tid * 65;                          // tid*64 + pad(tid)
        for (int t = 0; t < 64; ++t) {
            run += cdf[base + t];
            cdf[base + t] = run;
        }
        totals[tid] = run;
    }
    __syncthreads();

    // Phase 2: wave 0 scans the 256 partials as 16x16 f32 matrix via WMMA:
    //   P = L * X,  L lower-triangular ones, contracted as 4 chained
    //   V_WMMA_F32_16X16X4_F32 (K = 16 split into 4 chunks of 4).
    if (tid < 32) {
        const int m  = tid & 15;      // A row / B-D column index
        const int hi = tid >> 4;      // lanes 16-31 carry K=2,3 per ISA layout
        v8f acc = {};
#pragma unroll
        for (int c = 0; c < 4; ++c) {
            int k0 = 4 * c + 2 * hi;
            v2f a, bm;
            a.x = (k0     <= m) ? 1.0f : 0.0f;        // L[m][k0]
            a.y = (k0 + 1 <= m) ? 1.0f : 0.0f;        // L[m][k0+1]
            bm.x = totals[16 * m + k0];               // X[k0][m]
            bm.y = totals[16 * m + k0 + 1];           // X[k0+1][m]
            acc = __builtin_amdgcn_wmma_f32_16x16x4_f32(
                false, a, false, bm, (short)0, acc, false, false);
        }
        // acc[v] = prefix at offset v+8*hi within segment m
        float top = acc[7];                           // lane 16+m holds seg total
        float tot = __shfl(top, 16 + m, 32);          // my segment's total
        float sc = tot;                               // scan totals over m (width 16)
#pragma unroll
        for (int d = 1; d < 16; d <<= 1) {
            float o = __shfl_up(sc, d, 16);
            if (m >= d) sc += o;
        }
        float segoff = sc - tot;                      // exclusive segment offset
#pragma unroll
        for (int v = 0; v < 8; ++v)
            totsc[16 * m + 8 * hi + v] = acc[v] + segoff;  // inclusive over 256
        if (tid == 31) {
            float T = acc[7] + segoff;                // grand total (~1.0)
            sT[0] = T;
            sT[1] = __logf(T);
        }
    }
    __syncthreads();

    // Phase 3: add exclusive offsets back into the full CDF
    {
        float off = tid ? totsc[tid - 1] : 0.0f;
        int base = tid * 65;
        for (int t = 0; t < 64; ++t) cdf[base + t] += off;
    }
    __syncthreads();

    const float T = sT[0], logT = sT[1];
    const float* srow = states + (size_t)b * NN * 3;
    const float* wrow = pw + (size_t)b * NN;
    float* os = out_states + (size_t)b * NN * 3;
    float* ow = out_w + (size_t)b * NN;

    // Phase 4: threefry -> u, binary search LDS CDF, gather + corrected weight
    for (int t = 0; t < NN / TPB; ++t) {
        int i = t * TPB + tid;
        unsigned fidx = (unsigned)b * (unsigned)NN + (unsigned)i;
        bool firstHalf = fidx < HALF_COUNT;
        unsigned c0 = firstHalf ? fidx : fidx - HALF_COUNT;
        unsigned bits = threefry2x32_k01(c0, c0 + HALF_COUNT, firstHalf);
        float u = __uint_as_float((bits >> 9) | 0x3f800000u) - 1.0f;
        float target = u * T;                         // search un-normalized CDF

        int lo = 0, hiN = NN;
        while (lo < hiN) {                            // lower_bound, 14-15 LDS hits
            int mid = (lo + hiN) >> 1;
            float v = cdf[mid + (mid >> 6)];
            if (v < target) lo = mid + 1; else hiN = mid;
        }
        int idx = lo < NN ? lo : NN - 1;

        const float* sp = srow + 3 * (size_t)idx;     // gather (L2-resident row)
        float s0 = sp[0], s1 = sp[1], s2 = sp[2];
        size_t o3 = 3 * (size_t)i;
        os[o3 + 0] = s0; os[o3 + 1] = s1; os[o3 + 2] = s2;

        float wj = wrow[idx];
        float pj = ALPHA_F * __expf(wj - lse) + CU;   // recompute q at idx
        ow[i] = (wj - lse) - __logf(pj) + logT;       // new_w = pw_n - q_n
    }
}

extern "C" void kernel_launch(void* const* d_in, const int* in_sizes, int n_in,
                              void* d_out, int out_size, void* d_ws, size_t ws_size,
                              hipStream_t stream) {
    const float* states = (const float*)d_in[0];   // [B, N, 3]
    const float* pw     = (const float*)d_in[1];   // [B, N]
    float* out_states = (float*)d_out;                         // [B, N, 3]
    float* out_w      = out_states + (size_t)BB * NN * 3;      // [B, N]
    float* lse        = (float*)d_ws;                          // B floats scratch

    size_t shmem = (size_t)(PADN + TPB + TPB + 2) * sizeof(float);  // ~68 KB
    (void)hipFuncSetAttribute((const void*)k_resample,
                              hipFuncAttributeMaxDynamicSharedMemorySize,
                              (int)shmem);

    k_row_lse<<<BB, TPB, 0, stream>>>(pw, lse);
    k_resample<<<BB, TPB, shmem, stream>>>(states, pw, lse, out_states, out_w);
}